// ES_40072044872205
// MI455X (gfx1250) — compile-verified
//
#include <hip/hip_runtime.h>

// Problem constants (from reference)
#define HORIZON 24
#define MM      24
#define BB      16384
#define TT      2048
#define NN      (TT + BB - 1)      // 18431 scan steps
#define NPAD    18440              // NN rounded up to multiple of 8 (keeps float4 alignment)
#define NCHUNK  ((NN + MM - 1) / MM)   // 768 chunks of 24

// rcp with one Newton-Raphson refinement: ~full fp32 accuracy, 3 VALU ops,
// and (crucially) pipelined throughput instead of the long IEEE-div sequence.
__device__ __forceinline__ float frcp_nr(float x) {
    float r = __builtin_amdgcn_rcpf(x);
    r = __builtin_fmaf(__builtin_fmaf(-x, r, 1.0f), r, r);
    return r;
}

// ---------------------------------------------------------------------------
// Kernel 0: gather the scan input sequence ts[t]:
//   t <  T : x[0, t, 0]            = x[t]
//   t >= T : x[t-T+1, T-1, 0]      = x[(t-T+2)*T - 1]
// ---------------------------------------------------------------------------
__global__ void es_gather_ts(const float* __restrict__ x, float* __restrict__ ts) {
    int t = blockIdx.x * blockDim.x + threadIdx.x;
    if (t >= NN) return;
    ts[t] = (t < TT) ? x[t] : x[(t - TT + 2) * TT - 1];
}

// ---------------------------------------------------------------------------
// Kernel 1: the sequential Holt-Winters scan. One wave, lane 0.
// Ring buffer q[24] is kept in VGPRs via 24-step chunking with static indices.
// Per chunk: all 24 reciprocals of q[] and c[j]=a*y*rs are off the critical
// path (s_t has 24 steps of slack); the l-chain is one FMA per step.
// Emits compact arrays: xns[t], S[t] (=s_new at step t), Lev[t-(T-1)].
// ---------------------------------------------------------------------------
__global__ void es_scan(const float* __restrict__ ts,
                        const float* __restrict__ alpha_p,
                        const float* __restrict__ gamma_p,
                        const float* __restrict__ init_seas,
                        const float* __restrict__ level_p,
                        float* __restrict__ xns,
                        float* __restrict__ S,
                        float* __restrict__ Lev) {
    if (threadIdx.x != 0) return;

    const float a     = alpha_p[0];
    const float g     = gamma_p[0];
    const float one_a = 1.0f - a;
    const float one_g = 1.0f - g;
    float l = level_p[0];

    float q[MM];
#pragma unroll
    for (int j = 0; j < MM; ++j) q[j] = init_seas[j];

    for (int c = 0; c < NCHUNK; ++c) {
        const int base = c * MM;

        // Load 24 inputs as 6 x float4 (base is a multiple of 24 floats -> 16B aligned).
        float y[MM];
        const float4* yv = (const float4*)(ts + base);
#pragma unroll
        for (int k = 0; k < 6; ++k) {
            float4 v = yv[k];
            y[4 * k + 0] = v.x; y[4 * k + 1] = v.y;
            y[4 * k + 2] = v.z; y[4 * k + 3] = v.w;
        }

        // Off-chain work: reciprocals of the whole ring, and c[j] = a*y*rs.
        float rs[MM], yrs[MM], cc[MM];
#pragma unroll
        for (int j = 0; j < MM; ++j) rs[j] = frcp_nr(q[j]);
#pragma unroll
        for (int j = 0; j < MM; ++j) { yrs[j] = y[j] * rs[j]; cc[j] = a * yrs[j]; }

        // Critical chain: one FMA per step.
        float lt[MM];
#pragma unroll
        for (int j = 0; j < MM; ++j) {
            l = __builtin_fmaf(one_a, l, cc[j]);
            lt[j] = l;
        }

        // Off-chain epilogue: seasonality update, normalized output.
        float xnv[MM], snv[MM];
#pragma unroll
        for (int j = 0; j < MM; ++j) {
            float rl   = frcp_nr(lt[j]);
            float snew = __builtin_fmaf(one_g, q[j], g * y[j] * rl);
            q[j]   = snew;
            snv[j] = snew;
            xnv[j] = yrs[j] * rl;   // y/(s*l) = (y/s)*(1/l)
        }

        // Vector stores (padding region absorbs the 1 extra element of the last chunk).
        float4* xo = (float4*)(xns + base);
        float4* so = (float4*)(S + base);
#pragma unroll
        for (int k = 0; k < 6; ++k) {
            xo[k] = make_float4(xnv[4 * k], xnv[4 * k + 1], xnv[4 * k + 2], xnv[4 * k + 3]);
            so[k] = make_float4(snv[4 * k], snv[4 * k + 1], snv[4 * k + 2], snv[4 * k + 3]);
        }

        // Level snapshots for the denorm output (guarded scalar stores).
#pragma unroll
        for (int j = 0; j < MM; ++j) {
            int t = base + j;
            if (t >= TT - 1 && t < NN) Lev[t - (TT - 1)] = lt[j];
        }
    }
}

// ---------------------------------------------------------------------------
// Kernel 2: banded broadcast x_out[b, t] = xns[b + t].
// One block per row b; 512 threads x float4 = 2048 floats. Store-BW bound
// (134 MB); source (72 KB) is L2-resident. Prefetch keeps the read stream warm.
// ---------------------------------------------------------------------------
__global__ void es_fill_out(const float* __restrict__ xns, float* __restrict__ xout) {
    const int b  = blockIdx.x;
    const int t0 = threadIdx.x << 2;
    const float* src = xns + b + t0;
    __builtin_prefetch(src + 512, 0, 1);   // global_prefetch_b8
    float4 v = make_float4(src[0], src[1], src[2], src[3]);
    *(float4*)(xout + (size_t)b * TT + t0) = v;
}

// ---------------------------------------------------------------------------
// Kernel 3: denorm[b, h, 0] = Lev[b]; denorm[b, h, 1] = S[(T-24) + b + h].
// ---------------------------------------------------------------------------
__global__ void es_fill_denorm(const float* __restrict__ S,
                               const float* __restrict__ Lev,
                               float* __restrict__ out2) {
    int i = blockIdx.x * blockDim.x + threadIdx.x;   // i = b*24 + h
    if (i >= BB * HORIZON) return;
    int b = i / HORIZON;
    int h = i - b * HORIZON;
    float2 v = make_float2(Lev[b], S[(TT - HORIZON) + b + h]);
    *(float2*)(out2 + (size_t)i * 2) = v;
}

extern "C" void kernel_launch(void* const* d_in, const int* in_sizes, int n_in,
                              void* d_out, int out_size, void* d_ws, size_t ws_size,
                              hipStream_t stream) {
    const float* x         = (const float*)d_in[0];
    const float* alpha     = (const float*)d_in[1];
    const float* gam       = (const float*)d_in[2];
    const float* init_seas = (const float*)d_in[3];
    const float* level     = (const float*)d_in[4];

    float* out   = (float*)d_out;
    float* ws    = (float*)d_ws;
    float* ts    = ws;                 // [NPAD]
    float* xns   = ws + NPAD;          // [NPAD]
    float* S     = ws + 2 * NPAD;      // [NPAD]
    float* Lev   = ws + 3 * NPAD;      // [BB]
    // total ws use: (3*18440 + 16384) * 4 B ~= 287 KB

    es_gather_ts<<<(NN + 255) / 256, 256, 0, stream>>>(x, ts);
    es_scan<<<1, 32, 0, stream>>>(ts, alpha, gam, init_seas, level, xns, S, Lev);
    es_fill_out<<<BB, 512, 0, stream>>>(xns, out);
    es_fill_denorm<<<(BB * HORIZON + 255) / 256, 256, 0, stream>>>(
        S, Lev, out + (size_t)BB * TT);
}